// MLFFNet_70652212019433
// MI455X (gfx1250) — compile-verified
//
#include <hip/hip_runtime.h>
#include <hip/hip_bf16.h>
#include <math.h>

typedef __attribute__((ext_vector_type(2))) float v2f;
typedef __attribute__((ext_vector_type(8))) float v8f;

#define B_SZ    32
#define NATOM   256
#define NN      100
#define NF      42      // features
#define NH      50      // hidden width (both layers)
#define PADROWS 257     // dE_pad rows per batch (row 0 = zeros)
#define LDH     56      // LDS row stride (pad vs bank conflicts)

// D = A(16x4 f32) * B(4x16 f32) + C(16x16 f32)
#define WMMA_K4(A, Bf, C) \
  __builtin_amdgcn_wmma_f32_16x16x4_f32(false, (A), false, (Bf), (short)0, (C), false, false)

// ---------------------------------------------------------------------------
// Zero the pad row (neighbor index 0) of the gradient table: gradPad[b][0][:]=0
// ---------------------------------------------------------------------------
__global__ void zero_pad_kernel(float* __restrict__ gradPad) {
  int i = blockIdx.x * blockDim.x + threadIdx.x;
  if (i < B_SZ * NF) {
    int b = i / NF, f = i - b * NF;
    gradPad[(size_t)b * PADROWS * NF + f] = 0.f;
  }
}

// ---------------------------------------------------------------------------
// MLP forward + input-gradient, 16 atoms per wave32, fp32 WMMA (16x16x4).
// grid = 2 types * 32 batches * 8 tiles = 512 waves.
// Writes Ei (B*256) and gradPad rows 1..256 (B,257,42).
// ---------------------------------------------------------------------------
__global__ __launch_bounds__(32) void mlp_kernel(
    const float* __restrict__ image,
    const float* __restrict__ W0a, const float* __restrict__ b0a,
    const float* __restrict__ W1a, const float* __restrict__ b1a,
    const float* __restrict__ W2a, const float* __restrict__ b2a,
    const float* __restrict__ W0b, const float* __restrict__ b0b,
    const float* __restrict__ W1b, const float* __restrict__ b1b,
    const float* __restrict__ W2b, const float* __restrict__ b2b,
    float* __restrict__ Ei, float* __restrict__ gradPad)
{
  const int lane = threadIdx.x & 31;
  const int half = lane >> 4;        // 0: lanes 0-15, 1: lanes 16-31
  const int r    = lane & 15;

  const int id   = blockIdx.x;       // 0..511
  const int t    = id >> 8;          // atom type
  const int rem  = id & 255;
  const int b    = rem >> 3;         // batch
  const int tile = rem & 7;          // 16-atom tile inside the 128-atom type block

  const float* W0 = t ? W0b : W0a;
  const float* b0 = t ? b0b : b0a;
  const float* W1 = t ? W1b : W1a;
  const float* b1 = t ? b1b : b1a;
  const float* W2 = t ? W2b : W2a;
  const float* b2 = t ? b2b : b2a;

  __shared__ float Hs[16][LDH];      // staging tile for layout transposes

  // B-fragment loader: logical B is K x N. trans=false: B[k][n]=M[k*ld+n]
  // trans=true:  B[k][n]=M[n*ld+k] (i.e. M^T with M stored N x K row-major)
  auto loadB = [&](const float* M, int ld, int K, int N, int k0, int n0, bool trans) -> v2f {
    int n = n0 + r;
    int k = k0 + (half << 1);
    float e0 = 0.f, e1 = 0.f;
    if (n < N) {
      if (k     < K) e0 = trans ? M[n * ld + k]     : M[k * ld + n];
      if (k + 1 < K) e1 = trans ? M[n * ld + k + 1] : M[(k + 1) * ld + n];
    }
    v2f bf; bf[0] = e0; bf[1] = e1;
    return bf;
  };

  // ---- A-fragments of X (16 x 42, K padded to 44) ----
  v2f aX[11];
  {
    const float* xrow = image + (size_t)(b * NATOM + t * 128 + tile * 16 + r) * NF;
#pragma unroll
    for (int kk = 0; kk < 11; ++kk) {
      int k = kk * 4 + (half << 1);
      v2f a;
      a[0] = (k     < NF) ? xrow[k]     : 0.f;
      a[1] = (k + 1 < NF) ? xrow[k + 1] : 0.f;
      aX[kk] = a;
    }
  }

  // ---- layer 0: Z0 = X * W0  (N padded to 64) ----
  v8f z0[4];
#pragma unroll
  for (int nt = 0; nt < 4; ++nt) {
    v8f c = {};
#pragma unroll
    for (int kk = 0; kk < 11; ++kk) {
      v2f bf = loadB(W0, NH, NF, NH, kk * 4, nt * 16, false);
      c = WMMA_K4(aX[kk], bf, c);
    }
    z0[nt] = c;
  }

  // h0 = tanh(z0 + b0); keep D-layout copy and stage (16 x 52) to LDS
  v8f h0[4];
#pragma unroll
  for (int nt = 0; nt < 4; ++nt) {
    int col = nt * 16 + r;
    float bias = (col < NH) ? b0[col] : 0.f;
#pragma unroll
    for (int j = 0; j < 8; ++j) {
      float v = (col < NH) ? tanhf(z0[nt][j] + bias) : 0.f;
      h0[nt][j] = v;
      if (col < 52) Hs[j + 8 * half][col] = v;
    }
  }
  __syncthreads();

  v2f aH[13];
#pragma unroll
  for (int kk = 0; kk < 13; ++kk) {
    int k = kk * 4 + (half << 1);
    v2f a; a[0] = Hs[r][k]; a[1] = Hs[r][k + 1];
    aH[kk] = a;
  }

  // ---- layer 1: Z1 = H0 * W1  (K padded to 52) ----
  v8f z1[4];
#pragma unroll
  for (int nt = 0; nt < 4; ++nt) {
    v8f c = {};
#pragma unroll
    for (int kk = 0; kk < 13; ++kk) {
      v2f bf = loadB(W1, NH, NH, NH, kk * 4, nt * 16, false);
      c = WMMA_K4(aH[kk], bf, c);
    }
    z1[nt] = c;
  }

  // h1 = tanh(z1 + b1); Ei = h1 . W2 + b2 via cross-lane reduction
  float p[8];
#pragma unroll
  for (int j = 0; j < 8; ++j) p[j] = 0.f;
  v8f h1[4];
#pragma unroll
  for (int nt = 0; nt < 4; ++nt) {
    int col = nt * 16 + r;
    float bias = (col < NH) ? b1[col] : 0.f;
    float w2   = (col < NH) ? W2[col] : 0.f;
#pragma unroll
    for (int j = 0; j < 8; ++j) {
      float v = (col < NH) ? tanhf(z1[nt][j] + bias) : 0.f;
      h1[nt][j] = v;
      p[j] += v * w2;
    }
  }
#pragma unroll
  for (int j = 0; j < 8; ++j) {
    float s = p[j];
    s += __shfl_xor(s, 1, 32);
    s += __shfl_xor(s, 2, 32);
    s += __shfl_xor(s, 4, 32);
    s += __shfl_xor(s, 8, 32);                 // sum over the 16-lane half
    if (r == 0) {
      int atom = t * 128 + tile * 16 + j + 8 * half;
      Ei[b * NATOM + atom] = s + b2[0];
    }
  }

  // ---- backward: dz1 = (1-h1^2) * W2[col], stage ----
  __syncthreads();
#pragma unroll
  for (int nt = 0; nt < 4; ++nt) {
    int col = nt * 16 + r;
    float w2 = (col < NH) ? W2[col] : 0.f;
#pragma unroll
    for (int j = 0; j < 8; ++j) {
      float h = h1[nt][j];
      float v = (col < NH) ? (1.f - h * h) * w2 : 0.f;
      if (col < 52) Hs[j + 8 * half][col] = v;
    }
  }
  __syncthreads();
  v2f aZ[13];
#pragma unroll
  for (int kk = 0; kk < 13; ++kk) {
    int k = kk * 4 + (half << 1);
    v2f a; a[0] = Hs[r][k]; a[1] = Hs[r][k + 1];
    aZ[kk] = a;
  }

  // dh0 = dz1 * W1^T
  v8f dh0[4];
#pragma unroll
  for (int nt = 0; nt < 4; ++nt) {
    v8f c = {};
#pragma unroll
    for (int kk = 0; kk < 13; ++kk) {
      v2f bf = loadB(W1, NH, NH, NH, kk * 4, nt * 16, true);
      c = WMMA_K4(aZ[kk], bf, c);
    }
    dh0[nt] = c;
  }

  // dz0 = dh0 * (1-h0^2), stage
  __syncthreads();
#pragma unroll
  for (int nt = 0; nt < 4; ++nt) {
    int col = nt * 16 + r;
#pragma unroll
    for (int j = 0; j < 8; ++j) {
      float h = h0[nt][j];
      float v = (col < NH) ? dh0[nt][j] * (1.f - h * h) : 0.f;
      if (col < 52) Hs[j + 8 * half][col] = v;
    }
  }
  __syncthreads();
#pragma unroll
  for (int kk = 0; kk < 13; ++kk) {
    int k = kk * 4 + (half << 1);
    v2f a; a[0] = Hs[r][k]; a[1] = Hs[r][k + 1];
    aZ[kk] = a;
  }

  // dx = dz0 * W0^T  (K=50 padded 52, N=42 padded 48); write to gradPad rows 1..
#pragma unroll
  for (int nt = 0; nt < 3; ++nt) {
    v8f c = {};
#pragma unroll
    for (int kk = 0; kk < 13; ++kk) {
      v2f bf = loadB(W0, NH, NH, NF, kk * 4, nt * 16, true);
      c = WMMA_K4(aZ[kk], bf, c);
    }
    int col = nt * 16 + r;
    if (col < NF) {
#pragma unroll
      for (int j = 0; j < 8; ++j) {
        int atom = t * 128 + tile * 16 + j + 8 * half;
        gradPad[((size_t)b * PADROWS + 1 + atom) * NF + col] = c[j];
      }
    }
  }
}

// ---------------------------------------------------------------------------
// Etot[b] = sum_n Ei[b,n]  (deterministic tree reduction, no atomics)
// ---------------------------------------------------------------------------
__global__ __launch_bounds__(256) void etot_kernel(const float* __restrict__ Ei,
                                                   float* __restrict__ Etot) {
  int b = blockIdx.x;
  float v = Ei[b * NATOM + threadIdx.x];
#pragma unroll
  for (int m = 16; m >= 1; m >>= 1) v += __shfl_xor(v, m, 32);
  __shared__ float ws[8];
  if ((threadIdx.x & 31) == 0) ws[threadIdx.x >> 5] = v;
  __syncthreads();
  if (threadIdx.x == 0) {
    float s = 0.f;
#pragma unroll
    for (int i = 0; i < 8; ++i) s += ws[i];
    Etot[b] = s;
  }
}

// ---------------------------------------------------------------------------
// Force[b,n,:] = sum_{k,f} dE_pad[b, nbr[b,n,k], f] * dfeat[b,n,k,f,:]
// One 256-thread block per (b,n); streams 50.4 KB of dfeat contiguously.
// ---------------------------------------------------------------------------
__global__ __launch_bounds__(256) void force_kernel(
    const float* __restrict__ dfeat, const int* __restrict__ neighbor,
    const float* __restrict__ gradPad, float* __restrict__ F) {
  int bn = blockIdx.x;               // b*256 + n
  int b  = bn >> 8;

  __shared__ int nbr[NN];
  for (int k = threadIdx.x; k < NN; k += blockDim.x)
    nbr[k] = neighbor[(size_t)bn * NN + k];
  __syncthreads();

  const float* df = dfeat + (size_t)bn * NN * NF * 3;
  const float* gb = gradPad + (size_t)b * PADROWS * NF;

  float fx = 0.f, fy = 0.f, fz = 0.f;
  for (int p = threadIdx.x; p < NN * NF; p += 256) {
    int k = p / NF, f = p - k * NF;
    float dE = gb[nbr[k] * NF + f];          // L2-resident gather (1.4 MB table)
    const float* d3 = df + (size_t)p * 3;    // contiguous union across threads
    fx = fmaf(dE, d3[0], fx);
    fy = fmaf(dE, d3[1], fy);
    fz = fmaf(dE, d3[2], fz);
  }
#pragma unroll
  for (int m = 16; m >= 1; m >>= 1) {
    fx += __shfl_xor(fx, m, 32);
    fy += __shfl_xor(fy, m, 32);
    fz += __shfl_xor(fz, m, 32);
  }
  __shared__ float wsum[8][3];
  if ((threadIdx.x & 31) == 0) {
    int w = threadIdx.x >> 5;
    wsum[w][0] = fx; wsum[w][1] = fy; wsum[w][2] = fz;
  }
  __syncthreads();
  if (threadIdx.x < 3) {
    float s = 0.f;
#pragma unroll
    for (int i = 0; i < 8; ++i) s += wsum[i][threadIdx.x];
    F[(size_t)bn * 3 + threadIdx.x] = s;
  }
}

// ---------------------------------------------------------------------------
extern "C" void kernel_launch(void* const* d_in, const int* in_sizes, int n_in,
                              void* d_out, int out_size, void* d_ws, size_t ws_size,
                              hipStream_t stream) {
  const float* image    = (const float*)d_in[0];   // (32,256,42)
  const float* dfeat    = (const float*)d_in[1];   // (32,256,100,42,3)
  const int*   neighbor = (const int*)  d_in[2];   // (32,256,100)
  const float* W0a = (const float*)d_in[3];
  const float* b0a = (const float*)d_in[4];
  const float* W1a = (const float*)d_in[5];
  const float* b1a = (const float*)d_in[6];
  const float* W2a = (const float*)d_in[7];
  const float* b2a = (const float*)d_in[8];
  const float* W0b = (const float*)d_in[9];
  const float* b0b = (const float*)d_in[10];
  const float* W1b = (const float*)d_in[11];
  const float* b1b = (const float*)d_in[12];
  const float* W2b = (const float*)d_in[13];
  const float* b2b = (const float*)d_in[14];

  float* out  = (float*)d_out;
  float* Etot = out;                       // 32
  float* Ei   = out + 32;                  // 32*256
  float* F    = out + 32 + B_SZ * NATOM;   // 32*256*3

  float* gradPad = (float*)d_ws;           // (32,257,42) fp32 = 1.38 MB

  zero_pad_kernel<<<(B_SZ * NF + 255) / 256, 256, 0, stream>>>(gradPad);
  mlp_kernel<<<512, 32, 0, stream>>>(image,
                                     W0a, b0a, W1a, b1a, W2a, b2a,
                                     W0b, b0b, W1b, b1b, W2b, b2b,
                                     Ei, gradPad);
  etot_kernel<<<B_SZ, 256, 0, stream>>>(Ei, Etot);
  force_kernel<<<B_SZ * NATOM, 256, 0, stream>>>(dfeat, neighbor, gradPad, F);
}